// GINGraphClassifier_20461224198763
// MI455X (gfx1250) — compile-verified
//
#include <hip/hip_runtime.h>
#include <hip/hip_bf16.h>

#define NN 100000            // N_NODES (6250 * 16 exactly)
#define NE 640000            // N_EDGES
#define NG 128               // NUM_GRAPHS
#define NF 128               // NUM_FEATURES
#define NH 16                // HIDDEN
#define NC 10                // NUM_CLASSES
#define N16 (NN * NH)        // 1,600,000
#define TILES (NN / 16)      // 6250

typedef __attribute__((ext_vector_type(2))) float v2f;
typedef __attribute__((ext_vector_type(8))) float v8f;

#if defined(__has_builtin)
#if __has_builtin(__builtin_amdgcn_wmma_f32_16x16x4_f32)
#define HAVE_WMMA_F32 1
#endif
#endif

// ---------------------------------------------------------------- zero
__global__ void k_zero(float* __restrict__ p, int n) {
  int i = blockIdx.x * blockDim.x + threadIdx.x;
  if (i < n) p[i] = 0.0f;
}

// ---------------------------------------------------------------- GEMM1: y[N,16] = x[N,128] @ W1[128,16]
// One wave per 16-row tile; K swept 4 at a time with V_WMMA_F32_16X16X4_F32.
__global__ __launch_bounds__(256) void k_gemm_xw1(const float* __restrict__ x,
                                                  const float* __restrict__ W1,
                                                  float* __restrict__ y) {
  int wave = (int)((blockIdx.x * blockDim.x + threadIdx.x) >> 5);
  if (wave >= TILES) return;                 // uniform per wave -> EXEC all-1s inside
  int lane = threadIdx.x & 31;
  int m    = lane & 15;                      // row within tile (A) / col (B,D)
  int kc   = (lane >> 4) << 1;               // K-pair selector: 0 or 2
  int row_base = wave * 16;
  const float* xr = x + (size_t)(row_base + m) * NF;

  v8f c = {0.f,0.f,0.f,0.f,0.f,0.f,0.f,0.f};
#if HAVE_WMMA_F32
  for (int k0 = 0; k0 < NF; k0 += 4) {
    v2f a, b;
    a.x = xr[k0 + kc];
    a.y = xr[k0 + kc + 1];
    b.x = W1[(k0 + kc) * NH + m];
    b.y = W1[(k0 + kc + 1) * NH + m];
    c = __builtin_amdgcn_wmma_f32_16x16x4_f32(false, a, false, b,
                                              (short)0, c, false, false);
  }
#else
  // scalar fallback: lane (m, half) computes rows m+8*half? keep simple: each lane does
  // the same D layout by direct dot products.
  for (int v = 0; v < 8; ++v) {
    int row = row_base + v + ((lane >> 4) << 3);
    float s = 0.f;
    for (int k = 0; k < NF; ++k) s += x[(size_t)row * NF + k] * W1[k * NH + m];
    c[v] = s;
  }
#endif
  // D layout: VGPR v, lane l -> row (v + 8*(l>>4)), col (l&15)
  float* yr = y + (size_t)(row_base + ((lane >> 4) << 3)) * NH + m;
#pragma unroll
  for (int v = 0; v < 8; ++v) yr[(size_t)v * NH] = c[v];
}

// ---------------------------------------------------------------- GEMM2: z[N,16] = h[N,16] @ W2[16,16]
__global__ __launch_bounds__(256) void k_gemm_hw2(const float* __restrict__ h,
                                                  const float* __restrict__ W2,
                                                  float* __restrict__ z) {
  int wave = (int)((blockIdx.x * blockDim.x + threadIdx.x) >> 5);
  if (wave >= TILES) return;
  int lane = threadIdx.x & 31;
  int m    = lane & 15;
  int kc   = (lane >> 4) << 1;
  int row_base = wave * 16;
  const float* hr = h + (size_t)(row_base + m) * NH;

  v8f c = {0.f,0.f,0.f,0.f,0.f,0.f,0.f,0.f};
#if HAVE_WMMA_F32
#pragma unroll
  for (int k0 = 0; k0 < NH; k0 += 4) {
    v2f a, b;
    a.x = hr[k0 + kc];
    a.y = hr[k0 + kc + 1];
    b.x = W2[(k0 + kc) * NH + m];
    b.y = W2[(k0 + kc + 1) * NH + m];
    c = __builtin_amdgcn_wmma_f32_16x16x4_f32(false, a, false, b,
                                              (short)0, c, false, false);
  }
#else
  for (int v = 0; v < 8; ++v) {
    int row = row_base + v + ((lane >> 4) << 3);
    float s = 0.f;
    for (int k = 0; k < NH; ++k) s += h[(size_t)row * NH + k] * W2[k * NH + m];
    c[v] = s;
  }
#endif
  float* zr = z + (size_t)(row_base + ((lane >> 4) << 3)) * NH + m;
#pragma unroll
  for (int v = 0; v < 8; ++v) zr[(size_t)v * NH] = c[v];
}

// ---------------------------------------------------------------- edge scatter: agg[dst] += val[src]
// 16 threads per edge -> 64B-coalesced gathers and atomic groups; agg (6.4 MB) is L2-resident.
__global__ void k_scatter(const int* __restrict__ src, const int* __restrict__ dst,
                          const float* __restrict__ val, float* __restrict__ agg) {
  int idx = blockIdx.x * blockDim.x + threadIdx.x;
  int e = idx >> 4;
  if (e >= NE) return;
  int f = idx & 15;
  int s = src[e];
  int d = dst[e];
  atomicAdd(&agg[(size_t)d * NH + f], val[(size_t)s * NH + f]);
}

// ---------------------------------------------------------------- h = relu(y + agg + b)
__global__ void k_relu_bias(const float* __restrict__ y, const float* __restrict__ agg,
                            const float* __restrict__ b, float* __restrict__ h) {
  int i = blockIdx.x * blockDim.x + threadIdx.x;
  if (i >= N16) return;
  float v = y[i] + agg[i] + b[i & 15];
  h[i] = v > 0.f ? v : 0.f;
}

// ---------------------------------------------------------------- fused relu+bias + mean-pool accumulate
// batch is sorted -> run-length accumulate per thread, flush one atomic per (graph,chunk).
#define POOL_CHUNK 32
__global__ void k_pool(const float* __restrict__ z, const float* __restrict__ agg,
                       const float* __restrict__ b2, const int* __restrict__ batch,
                       float* __restrict__ psum, float* __restrict__ pcnt) {
  int t = blockIdx.x * blockDim.x + threadIdx.x;
  int f = t & 15;
  int n0 = (t >> 4) * POOL_CHUNK;
  if (n0 >= NN) return;
  int n1 = n0 + POOL_CHUNK; if (n1 > NN) n1 = NN;
  int curg = batch[n0];
  float acc = 0.f;
  int cnt = 0;
  for (int n = n0; n < n1; ++n) {
    int g = batch[n];
    if (g != curg) {
      atomicAdd(&psum[curg * NH + f], acc);
      if (f == 0) atomicAdd(&pcnt[curg], (float)cnt);
      acc = 0.f; cnt = 0; curg = g;
    }
    size_t i = (size_t)n * NH + f;
    float v = z[i] + agg[i] + b2[f];
    acc += (v > 0.f ? v : 0.f);
    cnt++;
  }
  atomicAdd(&psum[curg * NH + f], acc);
  if (f == 0) atomicAdd(&pcnt[curg], (float)cnt);
}

// ---------------------------------------------------------------- out[g,c] = (psum[g]/max(cnt,1)) @ Wfc + bfc
__global__ void k_outfc(const float* __restrict__ psum, const float* __restrict__ pcnt,
                        const float* __restrict__ Wfc, const float* __restrict__ bfc,
                        float* __restrict__ out) {
  int t = blockIdx.x * blockDim.x + threadIdx.x;
  if (t >= NG * NC) return;
  int g = t / NC, c = t - g * NC;
  float cnt = pcnt[g];
  cnt = cnt > 1.f ? cnt : 1.f;
  float inv = 1.f / cnt;
  float s = bfc[c];
#pragma unroll
  for (int f = 0; f < NH; ++f) s += (psum[g * NH + f] * inv) * Wfc[f * NC + c];
  out[t] = s;
}

// ----------------------------------------------------------------
extern "C" void kernel_launch(void* const* d_in, const int* in_sizes, int n_in,
                              void* d_out, int out_size, void* d_ws, size_t ws_size,
                              hipStream_t stream) {
  const float* x    = (const float*)d_in[0];
  const int*   edge = (const int*)d_in[1];     // [2, NE]
  const int*   batch= (const int*)d_in[2];     // [NN], sorted
  const float* W1   = (const float*)d_in[3];
  const float* b1   = (const float*)d_in[4];
  const float* W2   = (const float*)d_in[5];
  const float* b2   = (const float*)d_in[6];
  const float* Wfc  = (const float*)d_in[7];
  const float* bfc  = (const float*)d_in[8];
  float* out = (float*)d_out;
  float* ws  = (float*)d_ws;

  float* buf0 = ws;                            // y, then z      (6.4 MB)
  float* buf1 = ws + (size_t)N16;              // agg1 / agg2    (6.4 MB)
  float* buf2 = ws + 2 * (size_t)N16;          // h1             (6.4 MB)
  float* psum = ws + 3 * (size_t)N16;          // [NG*NH]
  float* pcnt = psum + NG * NH;                // [NG]

  const int* src = edge;
  const int* dst = edge + NE;

  const int B = 256;
  int zero1_n = N16 + NG * NH + NG;            // agg + pooled region (buf2 overwritten anyway)

  // 1) zero agg1 + pooled accumulators
  k_zero<<<(zero1_n + B - 1) / B, B, 0, stream>>>(buf1, N16);
  k_zero<<<(NG * NH + NG + B - 1) / B, B, 0, stream>>>(psum, NG * NH + NG);

  // 2) y = x @ W1   (WMMA f32 16x16x4, K=128)
  int gemm_blocks = (TILES + (B / 32) - 1) / (B / 32);
  k_gemm_xw1<<<gemm_blocks, B, 0, stream>>>(x, W1, buf0);

  // 3) agg1[dst] += y[src]
  k_scatter<<<(NE * 16 + B - 1) / B, B, 0, stream>>>(src, dst, buf0, buf1);

  // 4) h1 = relu(y + agg1 + b1)
  k_relu_bias<<<(N16 + B - 1) / B, B, 0, stream>>>(buf0, buf1, b1, buf2);

  // 5) re-zero agg
  k_zero<<<(N16 + B - 1) / B, B, 0, stream>>>(buf1, N16);

  // 6) z = h1 @ W2  (WMMA f32 16x16x4, K=16)
  k_gemm_hw2<<<gemm_blocks, B, 0, stream>>>(buf2, W2, buf0);

  // 7) agg2[dst] += z[src]
  k_scatter<<<(NE * 16 + B - 1) / B, B, 0, stream>>>(src, dst, buf0, buf1);

  // 8) pooled sums/counts with fused relu(z + agg2 + b2)
  int pool_threads = ((NN + POOL_CHUNK - 1) / POOL_CHUNK) * 16;
  k_pool<<<(pool_threads + B - 1) / B, B, 0, stream>>>(buf0, buf1, b2, batch, psum, pcnt);

  // 9) final FC
  k_outfc<<<(NG * NC + B - 1) / B, B, 0, stream>>>(psum, pcnt, Wfc, bfc, out);
}